// TFGraphNet_38053410242952
// MI455X (gfx1250) — compile-verified
//
#include <hip/hip_runtime.h>

// ---------------- problem constants (from reference) ----------------
#define N_NODES  50000
#define N_EDGES  800000
#define D_IN     16
#define D_HID    64
#define D_OUT    16
#define N_BLK    4

// tile geometry: one wave32 handles 16 edges (M=16), N=64 (4 WMMA N-tiles)
#define WAVES_PER_WG 8
#define TILES        (N_EDGES / 16)          // 50000, exact; 6250 blocks of 8 waves
#define LS           68                      // padded LDS row stride (floats): bank-conflict free

typedef __attribute__((ext_vector_type(2))) float v2f;
typedef __attribute__((ext_vector_type(8))) float v8f;

__device__ __forceinline__ v8f wmma4(v2f a, v2f b, v8f c) {
    return __builtin_amdgcn_wmma_f32_16x16x4_f32(false, a, false, b, (short)0, c, false, false);
}

// ---------------------------------------------------------------
// 16 K-steps (K=64) with A gathered directly from a per-lane global row.
// A-fragment (V_WMMA_F32_16X16X4_F32): lane L -> row M=L&15,
//   K = ks*4 + (L>=16?2:0) .. +1  => per-lane b64 from hrow + kb.
// B from prepacked weights: Wp[kb*64 + 2n + {0,1}] = W[kb..kb+1][n]
//   => one b64 per fragment.
// ---------------------------------------------------------------
__device__ __forceinline__ void mma_gather(const float* __restrict__ hrow,
                                           const float* __restrict__ Wp,
                                           int lane, v8f acc[4]) {
    const int koff  = (lane >> 4) * 2;
    const int ncol2 = (lane & 15) * 2;
    #pragma unroll 4
    for (int ks = 0; ks < 16; ++ks) {
        const int kb = ks * 4 + koff;
        v2f a = *(const v2f*)(hrow + kb);
        const float* wp = Wp + kb * D_HID + ncol2;
        #pragma unroll
        for (int nt = 0; nt < 4; ++nt) {
            v2f b = *(const v2f*)(wp + nt * 32);
            acc[nt] = wmma4(a, b, acc[nt]);
        }
    }
}

// same, but A fragments come from the per-wave 16x64 LDS tile (row stride LS)
__device__ __forceinline__ void mma_lds(const float* __restrict__ m,
                                        const float* __restrict__ Wp,
                                        int lane, v8f acc[4]) {
    const int koff  = (lane >> 4) * 2;
    const int mrow  = lane & 15;
    const int ncol2 = (lane & 15) * 2;
    #pragma unroll 4
    for (int ks = 0; ks < 16; ++ks) {
        const int kb = ks * 4 + koff;
        v2f a = *(const v2f*)(m + mrow * LS + kb);
        const float* wp = Wp + kb * D_HID + ncol2;
        #pragma unroll
        for (int nt = 0; nt < 4; ++nt) {
            v2f b = *(const v2f*)(wp + nt * 32);
            acc[nt] = wmma4(a, b, acc[nt]);
        }
    }
}

// bias + optional relu, write C/D-layout fragments to LDS (16x64, stride LS)
__device__ __forceinline__ void epilogue_to_lds(float* __restrict__ m,
                                                const float* __restrict__ Bv,
                                                int lane, v8f acc[4], bool relu) {
    const int half = lane >> 4;
    const int ncol = lane & 15;
    #pragma unroll
    for (int nt = 0; nt < 4; ++nt) {
        const int col  = nt * 16 + ncol;
        const float bi = Bv[col];
        #pragma unroll
        for (int v = 0; v < 8; ++v) {
            float val = acc[nt][v] + bi;                  // D element (M = v+8*half, N = col)
            if (relu) val = fmaxf(val, 0.0f);
            m[(v + half * 8) * LS + col] = val;
        }
    }
}

__global__ __launch_bounds__(32 * WAVES_PER_WG)
void edge_mlp_kernel(const float* __restrict__ h_in,
                     float* __restrict__ h_out,
                     const int* __restrict__ src,
                     const int* __restrict__ dst,
                     const float* __restrict__ Wp0, const float* __restrict__ B0,
                     const float* __restrict__ Wp1, const float* __restrict__ B1,
                     const float* __restrict__ Wp2, const float* __restrict__ B2) {
    __shared__ float lds[WAVES_PER_WG][16 * LS];          // 34.8 KB static
    const int lane = threadIdx.x & 31;
    const int wave = threadIdx.x >> 5;
    const int tile = blockIdx.x * WAVES_PER_WG + wave;    // grid sized so always < TILES
    float* m = lds[wave];

    const int half = lane >> 4;
    const int mrow = lane & 15;
    const int ncol = lane & 15;

    // each lane owns edge row M = lane&15 (duplicated across the two half-waves)
    const int e0     = tile * 16;
    const int my_src = src[e0 + mrow];
    const int my_dst = dst[e0 + mrow];

    v8f zf = {};
    v8f acc[4];

    // ---- layer 0: concat(h[src], h[dst]) [16x128] @ W0 [128x64], relu
    acc[0] = zf; acc[1] = zf; acc[2] = zf; acc[3] = zf;
    mma_gather(h_in + (size_t)my_src * D_HID, Wp0,                lane, acc);  // K = 0..63
    mma_gather(h_in + (size_t)my_dst * D_HID, Wp0 + 64 * D_HID,   lane, acc);  // K = 64..127
    epilogue_to_lds(m, B0, lane, acc, true);
    __syncthreads();

    // ---- layer 1: [16x64] @ W1 [64x64], relu
    acc[0] = zf; acc[1] = zf; acc[2] = zf; acc[3] = zf;
    mma_lds(m, Wp1, lane, acc);
    __syncthreads();                                      // all reads done before overwrite
    epilogue_to_lds(m, B1, lane, acc, true);
    __syncthreads();

    // ---- layer 2: [16x64] @ W2 [64x64], bias, scatter-add to h_out[dst]
    acc[0] = zf; acc[1] = zf; acc[2] = zf; acc[3] = zf;
    mma_lds(m, Wp2, lane, acc);
    #pragma unroll
    for (int nt = 0; nt < 4; ++nt) {
        const int col  = nt * 16 + ncol;
        const float bi = B2[col];
        #pragma unroll
        for (int v = 0; v < 8; ++v) {
            const int r  = v + half * 8;                  // row M of this D element
            const int nd = __shfl(my_dst, r, 32);         // lane r holds dst of edge r
            atomicAdd(h_out + (size_t)nd * D_HID + col, acc[nt][v] + bi);
        }
    }
}

// ---------------- weight prepack: pair rows (2k, 2k+1) element-interleaved ----------------
// out[p*128 + 2n + 0] = in[(2p)*64 + n];  out[p*128 + 2n + 1] = in[(2p+1)*64 + n]
__global__ void prepack_kernel(const float* __restrict__ in, float* __restrict__ out, int npairs) {
    int t = blockIdx.x * blockDim.x + threadIdx.x;
    if (t >= npairs * D_HID) return;
    int p = t >> 6, n = t & 63;
    out[p * 128 + 2 * n + 0] = in[(2 * p + 0) * D_HID + n];
    out[p * 128 + 2 * n + 1] = in[(2 * p + 1) * D_HID + n];
}

// ---------------- small VALU kernels ----------------
__global__ void lin_in_kernel(const float* __restrict__ x, const float* __restrict__ w,
                              const float* __restrict__ b, float* __restrict__ h) {
    int t = blockIdx.x * blockDim.x + threadIdx.x;
    if (t >= N_NODES * D_HID) return;
    int n = t >> 6, j = t & 63;
    const float* xr = x + (size_t)n * D_IN;
    float s = b[j];
    #pragma unroll
    for (int i = 0; i < D_IN; ++i) s += xr[i] * w[i * D_HID + j];
    h[t] = s;
}

__global__ void lin_out_kernel(const float* __restrict__ h, const float* __restrict__ w,
                               const float* __restrict__ b, float* __restrict__ out) {
    int t = blockIdx.x * blockDim.x + threadIdx.x;
    if (t >= N_NODES * D_OUT) return;
    int n = t >> 4, j = t & 15;
    const float* hr = h + (size_t)n * D_HID;
    float s = b[j];
    #pragma unroll
    for (int i = 0; i < D_HID; ++i) s += hr[i] * w[i * D_OUT + j];
    out[t] = s;
}

__global__ void zero_kernel(float* __restrict__ p, int n) {
    int t = blockIdx.x * blockDim.x + threadIdx.x;
    if (t < n) p[t] = 0.0f;
}

// ---------------- launcher ----------------
extern "C" void kernel_launch(void* const* d_in, const int* in_sizes, int n_in,
                              void* d_out, int out_size, void* d_ws, size_t ws_size,
                              hipStream_t stream) {
    const float* x        = (const float*)d_in[0];
    /* d_in[1] = lframes, unused by reference math */
    const int*   eidx     = (const int*)d_in[2];        // integer inputs delivered as int32
    const float* lin_in_w = (const float*)d_in[3];
    const float* lin_in_b = (const float*)d_in[4];
    const float* w0 = (const float*)d_in[5];
    const float* b0 = (const float*)d_in[6];
    const float* w1 = (const float*)d_in[7];
    const float* b1 = (const float*)d_in[8];
    const float* w2 = (const float*)d_in[9];
    const float* b2 = (const float*)d_in[10];
    const float* wo = (const float*)d_in[11];
    const float* bo = (const float*)d_in[12];

    const int* src = eidx;              // edge_index[0]
    const int* dst = eidx + N_EDGES;    // edge_index[1]

    // workspace layout: hA | hB | Wp0 | Wp1 | Wp2  (~25.9 MB total)
    float* hA  = (float*)d_ws;
    float* hB  = hA  + (size_t)N_NODES * D_HID;
    float* Wp0 = hB  + (size_t)N_NODES * D_HID;         // 4 * 128*64
    float* Wp1 = Wp0 + (size_t)N_BLK * 2 * D_HID * D_HID;
    float* Wp2 = Wp1 + (size_t)N_BLK * D_HID * D_HID;   // each 4 * 64*64

    // prepack all 4 blocks' weights (row counts are even, pairs never cross blocks)
    prepack_kernel<<<(N_BLK * 64 * D_HID + 255) / 256, 256, 0, stream>>>(w0, Wp0, N_BLK * 64);
    prepack_kernel<<<(N_BLK * 32 * D_HID + 255) / 256, 256, 0, stream>>>(w1, Wp1, N_BLK * 32);
    prepack_kernel<<<(N_BLK * 32 * D_HID + 255) / 256, 256, 0, stream>>>(w2, Wp2, N_BLK * 32);

    lin_in_kernel<<<(N_NODES * D_HID + 255) / 256, 256, 0, stream>>>(x, lin_in_w, lin_in_b, hA);

    float* hin = hA;
    float* hout = hB;
    for (int k = 0; k < N_BLK; ++k) {
        zero_kernel<<<(N_NODES * D_HID + 255) / 256, 256, 0, stream>>>(hout, N_NODES * D_HID);
        edge_mlp_kernel<<<TILES / WAVES_PER_WG, 32 * WAVES_PER_WG, 0, stream>>>(
            hin, hout, src, dst,
            Wp0 + (size_t)k * 2 * D_HID * D_HID, b0 + k * D_HID,
            Wp1 + (size_t)k * D_HID * D_HID,     b1 + k * D_HID,
            Wp2 + (size_t)k * D_HID * D_HID,     b2 + k * D_HID);
        float* t = hin; hin = hout; hout = t;
    }

    lin_out_kernel<<<(N_NODES * D_OUT + 255) / 256, 256, 0, stream>>>(hin, wo, bo, (float*)d_out);
}